// EfficientMemorySiLU_78821239816655
// MI455X (gfx1250) — compile-verified
//
#include <hip/hip_runtime.h>
#include <hip/hip_bf16.h>
#include <stdint.h>

// ---------------------------------------------------------------------------
// EfficientMemorySiLU for MI455X (gfx1250, wave32).
// Bandwidth-bound (~0.8 GB mandatory traffic -> ~35us floor @ 23.3 TB/s).
// L@R reconstruction tiles + randomized-SVD projection GEMMs use
// V_WMMA_F32_16X16X4_F32; L/R tiles are staged into LDS with the Tensor Data
// Mover (tensor_load_to_lds + s_wait_tensorcnt), and the 8-batch elementwise
// stream reuses each tile's WMMA result across all batches.
// ---------------------------------------------------------------------------

typedef __attribute__((ext_vector_type(2))) float v2f;
typedef __attribute__((ext_vector_type(8))) float v8f;
typedef __attribute__((ext_vector_type(4))) unsigned int u32x4;
typedef __attribute__((ext_vector_type(8))) int i32x8;
typedef __attribute__((ext_vector_type(4))) int i32x4;

#define NROW 2048
#define NCOL 2048
#define NBATCH 8
#define RANK 16
#define QMAXV 127.0f
#define HBINS 4096
#define HSCALE 512.0f // bin = |v|*512 covers [0, 8)

// ws layout (32-bit words)
#define WS_HIST 0       // 4096 uint
#define WS_THRESH 4096  // 1 float
#define WS_MAXBITS 4097 // 1 uint (bits of max |resid|)
#define WS_SCALE 4098   // 1 float
#define WS_NORMS 4112   // 16 float
#define WS_Y 4128       // 2048*16 float (projection, then normalized in place)
#define WS_ZERO_WORDS 4128

__device__ __forceinline__ float g_init(int k, int j) {
  // deterministic pseudo-random projection matrix G[2048][16]
  uint32_t h = (uint32_t)(k * RANK + j + 1) * 2654435761u;
  h ^= h >> 16; h *= 2246822519u; h ^= h >> 13;
  return ((float)(h & 0xFFFFu) * (1.0f / 32768.0f)) - 1.0f;
}

__device__ __forceinline__ float xsub0_at(const float* x0, int r, int c, float t) {
  float v = x0[r * NCOL + c];
  return (fabsf(v) > t) ? 0.0f : v;
}

__device__ __forceinline__ v8f wmma_k4(v8f c, v2f a, v2f b) {
  // D = A(16x4) * B(4x16) + C
  return __builtin_amdgcn_wmma_f32_16x16x4_f32(false, a, false, b, (short)0, c,
                                               false, false);
}

// --- Tensor Data Mover: stage a 16x16 f32 tile (row stride = `stride` elems)
// --- from global memory into LDS at byte offset `lds_addr`.
__device__ __forceinline__ void tdm_load_tile16x16_f32(const float* gsrc,
                                                       uint32_t lds_addr,
                                                       uint32_t tensor_w,
                                                       uint32_t tensor_h,
                                                       uint32_t stride) {
  const uint64_t ga = (uint64_t)(uintptr_t)gsrc;
  u32x4 g0;
  g0[0] = 1u;                                   // count=1, user descriptor
  g0[1] = lds_addr;                             // lds_addr (bytes)
  g0[2] = (uint32_t)ga;                         // global_addr[31:0]
  g0[3] = ((uint32_t)(ga >> 32) & 0x01FFFFFFu)  // global_addr[56:32]
          | (2u << 30);                         // type=2 ("image")
  i32x8 g1;
  g1[0] = (int)(2u << 16);                      // data_size=2 -> 4 bytes
  g1[1] = (int)((tensor_w & 0xFFFFu) << 16);    // tensor_dim0[15:0]
  g1[2] = (int)(((tensor_w >> 16) & 0xFFFFu) |  // tensor_dim0[31:16]
                ((tensor_h & 0xFFFFu) << 16));  // tensor_dim1[15:0]
  g1[3] = (int)(((tensor_h >> 16) & 0xFFFFu) |  // tensor_dim1[31:16]
                (16u << 16));                   // tile_dim0 = 16
  g1[4] = (int)16u;                             // tile_dim1 = 16, tile_dim2=0
  g1[5] = (int)stride;                          // tensor_dim0_stride[31:0]
  g1[6] = 0;                                    // stride[47:32] | dim1_stride lo
  g1[7] = 0;
  i32x4 z4 = {};
#if defined(__clang_major__) && (__clang_major__ >= 23)
  i32x8 z8 = {};
  __builtin_amdgcn_tensor_load_to_lds(g0, g1, z4, z4, z8, 0);
#else
  __builtin_amdgcn_tensor_load_to_lds(g0, g1, z4, z4, 0);
#endif
}

// --------------------------- phase 0: reset ws ------------------------------
__global__ void zero_ws_kernel(unsigned int* wsu) {
  int i = blockIdx.x * blockDim.x + threadIdx.x;
  if (i < WS_ZERO_WORDS) wsu[i] = 0u;
}

// --------------------- phase 1: quantile via histogram ----------------------
__global__ void hist_kernel(const float* __restrict__ x0,
                            unsigned int* __restrict__ hist) {
  const int n = NROW * NCOL;
  for (int i = blockIdx.x * blockDim.x + threadIdx.x; i < n;
       i += gridDim.x * blockDim.x) {
    float v = fabsf(x0[i]);
    int b = (int)(v * HSCALE);
    if (b > HBINS - 1) b = HBINS - 1;
    atomicAdd(&hist[b], 1u);
  }
}

__global__ void quantile_kernel(const unsigned int* __restrict__ hist,
                                float* __restrict__ wsf) {
  if (blockIdx.x == 0 && threadIdx.x == 0) {
    const unsigned long long target =
        (unsigned long long)(0.99 * (double)(NROW * NCOL));
    unsigned long long cum = 0;
    int bsel = HBINS - 1;
    for (int i = 0; i < HBINS; ++i) {
      cum += hist[i];
      if (cum >= target) { bsel = i; break; }
    }
    wsf[WS_THRESH] = ((float)bsel + 1.0f) / HSCALE;
  }
}

// ------------- phase 2: randomized rank-16 factorization (WMMA) -------------
// Y = X_sub[0] @ G   (2048x16).  One wave per 16-row tile.
__global__ void projY_kernel(const float* __restrict__ x0,
                             const float* __restrict__ wsf,
                             float* __restrict__ Y) {
  const int wave = (blockIdx.x * blockDim.x + threadIdx.x) >> 5;
  const int lane = threadIdx.x & 31;
  const int half = lane >> 4;  // 0: K pair {0,1}; 1: K pair {2,3}
  const int l15 = lane & 15;
  const int row0 = wave * 16;
  const float t = wsf[WS_THRESH];

  v8f c = {};
  for (int k0 = 0; k0 < NCOL; k0 += 4) {
    const int ka = k0 + 2 * half;
    v2f a, b;
    a.x = xsub0_at(x0, row0 + l15, ka, t);       // A: M=l15, K=ka
    a.y = xsub0_at(x0, row0 + l15, ka + 1, t);
    b.x = g_init(ka, l15);                       // B: K=ka, N=l15
    b.y = g_init(ka + 1, l15);
    c = wmma_k4(c, a, b);
  }
#pragma unroll
  for (int r = 0; r < 8; ++r) {
    const int m = r + 8 * half; // C/D: VGPR r -> M=r (lanes<16) / M=r+8
    Y[(row0 + m) * RANK + l15] = c[r];
  }
}

// Column norms of Y (deterministic tree reduction).
__global__ void colnorm_kernel(const float* __restrict__ Y,
                               float* __restrict__ wsf) {
  __shared__ float sbuf[256];
  const int col = blockIdx.x; // 0..15
  float s = 0.0f;
  for (int r = threadIdx.x; r < NROW; r += blockDim.x) {
    float v = Y[r * RANK + col];
    s += v * v;
  }
  sbuf[threadIdx.x] = s;
  __syncthreads();
  for (int off = 128; off > 0; off >>= 1) {
    if ((int)threadIdx.x < off) sbuf[threadIdx.x] += sbuf[threadIdx.x + off];
    __syncthreads();
  }
  if (threadIdx.x == 0) wsf[WS_NORMS + col] = sbuf[0];
}

// L = Y / ||col||  (write output L and normalize Y in place -> Q)
__global__ void writeL_kernel(float* __restrict__ Y,
                              const float* __restrict__ wsf,
                              float* __restrict__ Lout) {
  const int i = blockIdx.x * blockDim.x + threadIdx.x; // 2048*16
  const int col = i & (RANK - 1);
  const float inv = rsqrtf(fmaxf(wsf[WS_NORMS + col], 1e-20f));
  const float q = Y[i] * inv;
  Y[i] = q;
  Lout[i] = q;
}

// R = Q^T @ X_sub[0]   (16x2048).  One wave per 16-col tile.
__global__ void projR_kernel(const float* __restrict__ x0,
                             const float* __restrict__ wsf,
                             const float* __restrict__ Q,
                             float* __restrict__ Rout) {
  const int wave = (blockIdx.x * blockDim.x + threadIdx.x) >> 5;
  const int lane = threadIdx.x & 31;
  const int half = lane >> 4;
  const int l15 = lane & 15;
  const int col0 = wave * 16;
  const float t = wsf[WS_THRESH];

  v8f c = {};
  for (int k0 = 0; k0 < NROW; k0 += 4) {
    const int ka = k0 + 2 * half;
    v2f a, b;
    a.x = Q[ka * RANK + l15];              // A = Q^T: A[m][k] = Q[k][m]
    a.y = Q[(ka + 1) * RANK + l15];
    b.x = xsub0_at(x0, ka, col0 + l15, t); // B: K=ka, N=col0+l15
    b.y = xsub0_at(x0, ka + 1, col0 + l15, t);
    c = wmma_k4(c, a, b);
  }
#pragma unroll
  for (int r = 0; r < 8; ++r) {
    const int m = r + 8 * half;
    Rout[m * NCOL + col0 + l15] = c[r];
  }
}

// ---- phases 3/4 common: TDM-stage L/R tiles to LDS, WMMA the L@R tile ------
// One wave per 16x16 tile; 8 waves / 256-thread block; batch loop inside so
// the WMMA result and staged tiles are reused across all 8 batches.
__device__ __forceinline__ v8f lr_tile_from_lds(const float* Lt,
                                                const float* Rt, int half,
                                                int l15) {
  v8f c = {};
#pragma unroll
  for (int k0 = 0; k0 < RANK; k0 += 4) {
    const int ka = k0 + 2 * half;
    v2f a, bb;
    a.x = Lt[l15 * 16 + ka];        // A: M=l15, K=ka (tile is 16x16 row-major)
    a.y = Lt[l15 * 16 + ka + 1];
    bb.x = Rt[ka * 16 + l15];       // B: K=ka, N=l15
    bb.y = Rt[(ka + 1) * 16 + l15];
    c = wmma_k4(c, a, bb);
  }
  return c;
}

__global__ void __launch_bounds__(256)
silu_residmax_kernel(const float* __restrict__ x,
                     const float* __restrict__ wsf,
                     const float* __restrict__ L, const float* __restrict__ R,
                     float* __restrict__ result, float* __restrict__ xout,
                     unsigned int* __restrict__ maxbits) {
  __shared__ float smem[8 * 512]; // per wave: 256 floats Ltile + 256 Rtile
  const int wib = threadIdx.x >> 5;
  const int lane = threadIdx.x & 31;
  const int half = lane >> 4;
  const int l15 = lane & 15;
  const int waveId = blockIdx.x * 8 + wib; // 0..16383
  const int ty = waveId >> 7, tx = waveId & 127;
  const int row0 = ty * 16, col0 = tx * 16;
  const float t = wsf[WS_THRESH];

  float* Lt = &smem[wib * 512];
  float* Rt = Lt + 256;
  // TDM: L tile is 16 rows x 16 cols, row stride 16 (contiguous block);
  //      R tile is 16 rows x 16 cols, row stride 2048.
  tdm_load_tile16x16_f32(L + (size_t)row0 * RANK,
                         (uint32_t)(uintptr_t)Lt, RANK, NROW, RANK);
  tdm_load_tile16x16_f32(R + col0, (uint32_t)(uintptr_t)Rt, NCOL, RANK, NCOL);
  __builtin_amdgcn_s_wait_tensorcnt(0);

  const v8f c = lr_tile_from_lds(Lt, Rt, half, l15);

  float lmax = 0.0f;
  for (int b = 0; b < NBATCH; ++b) {
    const size_t base = (size_t)b * NROW * NCOL;
#pragma unroll
    for (int r = 0; r < 8; ++r) {
      const int m = r + 8 * half;
      const size_t idx = base + (size_t)(row0 + m) * NCOL + (col0 + l15);
      const float v = x[idx];
      const float sig = 1.0f / (1.0f + expf(-v));
      result[idx] = v * sig;
      const float xo = (fabsf(v) > t) ? v : 0.0f;
      xout[idx] = xo;
      lmax = fmaxf(lmax, fabsf((v - xo) - c[r]));
    }
  }
  atomicMax(maxbits, __float_as_uint(lmax)); // lmax>=0: bit order == fp order
}

__global__ void scale_kernel(const unsigned int* __restrict__ maxbits,
                             float* __restrict__ wsf,
                             float* __restrict__ scale_out) {
  if (blockIdx.x == 0 && threadIdx.x == 0) {
    float s = __uint_as_float(*maxbits) / QMAXV;
    s = fmaxf(s, 1e-20f);
    wsf[WS_SCALE] = s;
    *scale_out = s;
  }
}

__global__ void __launch_bounds__(256)
quant_kernel(const float* __restrict__ x, const float* __restrict__ wsf,
             const float* __restrict__ L, const float* __restrict__ R,
             float* __restrict__ qout) {
  __shared__ float smem[8 * 512];
  const int wib = threadIdx.x >> 5;
  const int lane = threadIdx.x & 31;
  const int half = lane >> 4;
  const int l15 = lane & 15;
  const int waveId = blockIdx.x * 8 + wib;
  const int ty = waveId >> 7, tx = waveId & 127;
  const int row0 = ty * 16, col0 = tx * 16;
  const float t = wsf[WS_THRESH];
  const float inv_scale = 1.0f / wsf[WS_SCALE];

  float* Lt = &smem[wib * 512];
  float* Rt = Lt + 256;
  tdm_load_tile16x16_f32(L + (size_t)row0 * RANK,
                         (uint32_t)(uintptr_t)Lt, RANK, NROW, RANK);
  tdm_load_tile16x16_f32(R + col0, (uint32_t)(uintptr_t)Rt, NCOL, RANK, NCOL);
  __builtin_amdgcn_s_wait_tensorcnt(0);

  const v8f c = lr_tile_from_lds(Lt, Rt, half, l15);

  for (int b = 0; b < NBATCH; ++b) {
    const size_t base = (size_t)b * NROW * NCOL;
#pragma unroll
    for (int r = 0; r < 8; ++r) {
      const int m = r + 8 * half;
      const size_t idx = base + (size_t)(row0 + m) * NCOL + (col0 + l15);
      const float v = x[idx];
      const float xo = (fabsf(v) > t) ? v : 0.0f;
      const float resid = (v - xo) - c[r];
      float q = rintf(resid * inv_scale);
      q = fminf(fmaxf(q, -QMAXV - 1.0f), QMAXV);
      qout[idx] = q;
    }
  }
}

// ---------------------------------------------------------------------------
extern "C" void kernel_launch(void* const* d_in, const int* in_sizes, int n_in,
                              void* d_out, int out_size, void* d_ws,
                              size_t ws_size, hipStream_t stream) {
  const float* x = (const float*)d_in[0];
  const float* x0 = x; // batch 0

  float* out = (float*)d_out;
  const size_t B = (size_t)NBATCH * NROW * NCOL;
  float* result = out;
  float* xout = out + B;
  float* qout = out + 2 * B;
  float* scale_out = out + 3 * B;
  float* Lout = out + 3 * B + 1;
  float* Rout = out + 3 * B + 1 + (size_t)NROW * RANK;

  float* wsf = (float*)d_ws;
  unsigned int* wsu = (unsigned int*)d_ws;
  unsigned int* hist = wsu + WS_HIST;
  unsigned int* maxbits = wsu + WS_MAXBITS;
  float* Y = wsf + WS_Y;

  // phase 0: reset accumulators
  zero_ws_kernel<<<(WS_ZERO_WORDS + 255) / 256, 256, 0, stream>>>(wsu);

  // phase 1: approximate 99th percentile of |x[0]|
  hist_kernel<<<1024, 256, 0, stream>>>(x0, hist);
  quantile_kernel<<<1, 32, 0, stream>>>(hist, wsf);

  // phase 2: randomized rank-16 factorization (WMMA GEMMs)
  projY_kernel<<<32, 128, 0, stream>>>(x0, wsf, Y);
  colnorm_kernel<<<RANK, 256, 0, stream>>>(Y, wsf);
  writeL_kernel<<<(NROW * RANK) / 256, 256, 0, stream>>>(Y, wsf, Lout);
  projR_kernel<<<32, 128, 0, stream>>>(x0, wsf, Y, Rout);

  // phase 3: fused SiLU + outlier split + |resid| max
  // 16384 waves (one per 16x16 tile), batches looped inside each wave
  silu_residmax_kernel<<<2048, 256, 0, stream>>>(x, wsf, Lout, Rout, result,
                                                 xout, maxbits);
  scale_kernel<<<1, 32, 0, stream>>>(maxbits, wsf, scale_out);

  // phase 4: quantize residual
  quant_kernel<<<2048, 256, 0, stream>>>(x, wsf, Lout, Rout, qout);
}